// ERPFeatureExtractor_18245021073948
// MI455X (gfx1250) — compile-verified
//
#include <hip/hip_runtime.h>
#include <math.h>

typedef __attribute__((ext_vector_type(2))) float v2f;
typedef __attribute__((ext_vector_type(4))) float v4f;
typedef __attribute__((ext_vector_type(8))) float v8f;

#define T_LEN 500
#define C_CH  128
#define P0    166
#define P1    333
#define N0    100
#define N1    175

__global__ __launch_bounds__(256) void erp_feat_kernel(const float* __restrict__ eeg,
                                                       float* __restrict__ out) {
    __shared__ float S[C_CH][16];   // per-channel quantities, padded to 16 cols
    __shared__ float red[20];       // [0..15] WMMA column sums, [16..18] extremes

    const int tid  = threadIdx.x;
    const int lane = tid & 31;
    const int wave = tid >> 5;           // 0..7
    const int b    = blockIdx.x;

    // ---------------- Stage 1: per-(b,c) time reductions (bandwidth bound) ----------------
    for (int i = 0; i < 16; ++i) {
        const int c = wave * 16 + i;
        const v4f* row = (const v4f*)(eeg + ((size_t)b * C_CH + c) * T_LEN);
        float pmax = -INFINITY; int pidx = 0x7fffffff;   // p300 max + first-index argmax
        float nmin =  INFINITY;                          // n2 min
        float asum = 0.f, bsum = 0.f;                    // alpha / beta sum of squares
        for (int v = lane; v < T_LEN / 4; v += 32) {     // 125 float4 per row, coalesced NT
            v4f x4 = __builtin_nontemporal_load(row + v);
            const int t0 = v * 4;
#pragma unroll
            for (int j = 0; j < 4; ++j) {
                float x  = x4[j];
                int   t  = t0 + j;
                float xs = x * x;
                if (t < P0)      { asum += xs; }
                else if (t < P1) { bsum += xs; if (x > pmax) { pmax = x; pidx = t; } }
                if (t >= N0 && t < N1) nmin = fminf(nmin, x);
            }
        }
        // wave32 butterfly reductions
#pragma unroll
        for (int off = 16; off > 0; off >>= 1) {
            float ov = __shfl_xor(pmax, off, 32);
            int   oi = __shfl_xor(pidx, off, 32);
            if (ov > pmax || (ov == pmax && oi < pidx)) { pmax = ov; pidx = oi; }
            nmin  = fminf(nmin, __shfl_xor(nmin, off, 32));
            asum += __shfl_xor(asum, off, 32);
            bsum += __shfl_xor(bsum, off, 32);
        }
        if (lane == 0) {
            float lat   = (float)pidx / 100.0f;          // absolute index / SFREQ
            float amp   = pmax;
            float alpha = asum / (float)P0;              // mean over 166
            float beta  = bsum / (float)(P1 - P0);       // mean over 167
            S[c][0] = lat;         S[c][1] = lat * lat;
            S[c][2] = amp;         S[c][3] = amp * lat;
            S[c][4] = nmin;        S[c][5] = nmin * nmin;
            S[c][6] = alpha;       S[c][7] = alpha * alpha;
            S[c][8] = beta;        S[c][9] = beta * beta;
#pragma unroll
            for (int q = 10; q < 16; ++q) S[c][q] = 0.f;
        }
    }
    __syncthreads();

    // ---------------- Stage 2: channel reductions; sums via V_WMMA_F32_16X16X4_F32 ----------------
    if (wave == 0) {   // full wave, EXEC all-ones as WMMA requires
        // extremes (not matmulable): each lane covers 4 channels
        float amax = -INFINITY, nmn = INFINITY, almax = -INFINITY;
#pragma unroll
        for (int i = 0; i < 4; ++i) {
            int c = lane + 32 * i;
            amax  = fmaxf(amax,  S[c][2]);
            nmn   = fminf(nmn,   S[c][4]);
            almax = fmaxf(almax, S[c][6]);
        }
#pragma unroll
        for (int off = 16; off > 0; off >>= 1) {
            amax  = fmaxf(amax,  __shfl_xor(amax,  off, 32));
            nmn   = fminf(nmn,   __shfl_xor(nmn,   off, 32));
            almax = fmaxf(almax, __shfl_xor(almax, off, 32));
        }

        // Column sums over 128 channels: D = ones(16x4) x B(4x16) + C, chained 32x.
        // B layout (4x16 f32): lanes 0-15 hold (K=0,N=lane) in v0, (K=1,N) in v1;
        //                      lanes 16-31 hold (K=2,N) in v0, (K=3,N) in v1.
        v2f a; a.x = 1.0f; a.y = 1.0f;          // all-ones A (16x4)
        v8f acc = {};
        const bool hi = lane >= 16;
        const int  n  = lane & 15;
        for (int k = 0; k < 32; ++k) {
            const int r0 = 4 * k + (hi ? 2 : 0);
            v2f bm;
            bm.x = S[r0][n];
            bm.y = S[r0 + 1][n];
            acc = __builtin_amdgcn_wmma_f32_16x16x4_f32(false, a, false, bm,
                                                        (short)0, acc, false, false);
        }
        if (lane < 16) red[lane] = acc[0];       // D[M=0][N=lane] = sum over 128 channels
        if (lane == 0) { red[16] = amax; red[17] = nmn; red[18] = almax; }
    }
    __syncthreads();

    // ---------------- Epilogue: 13 features ----------------
    if (tid == 0) {
        const float invN = 1.0f / 128.0f;
        const float invM = 1.0f / 127.0f;
        float s0 = red[0], s1 = red[1], s2 = red[2], s3 = red[3], s4 = red[4];
        float s5 = red[5], s6 = red[6], s7 = red[7], s8 = red[8], s9 = red[9];
        float* o = out + (size_t)b * 13;
        o[0]  = s0 * invN;                                                 // lat.mean
        o[1]  = sqrtf(fmaxf((s1 - s0 * s0 * invN) * invM, 0.0f));          // lat.std ddof=1
        o[2]  = s2 * invN;                                                 // amp.mean
        o[3]  = red[16];                                                   // amp.max
        o[4]  = s3 * invN;                                                 // (amp*lat).mean
        o[5]  = s4 * invN;                                                 // n2.mean
        o[6]  = sqrtf(fmaxf((s5 - s4 * s4 * invN) * invM, 0.0f));          // n2.std
        o[7]  = red[17];                                                   // n2.min
        o[8]  = s6 * invN;                                                 // alpha.mean
        o[9]  = sqrtf(fmaxf((s7 - s6 * s6 * invN) * invM, 0.0f));          // alpha.std
        o[10] = red[18];                                                   // alpha.max
        o[11] = s8 * invN;                                                 // beta.mean
        o[12] = sqrtf(fmaxf((s9 - s8 * s8 * invN) * invM, 0.0f));          // beta.std
    }
}

extern "C" void kernel_launch(void* const* d_in, const int* in_sizes, int n_in,
                              void* d_out, int out_size, void* d_ws, size_t ws_size,
                              hipStream_t stream) {
    const float* eeg = (const float*)d_in[0];
    float* out = (float*)d_out;
    const int B = in_sizes[0] / (C_CH * T_LEN);   // 2048
    erp_feat_kernel<<<B, 256, 0, stream>>>(eeg, out);
}